// RefineLoss_28467043238128
// MI455X (gfx1250) — compile-verified
//
#include <hip/hip_runtime.h>
#include <cstdint>

#define IMG_H 1024
#define IMG_W 1024
#define NIMG  16
#define TW    64      // tile width  (pixels per block, x)
#define TH    16      // tile height (pixels per block, y)
#define LW    76      // padded LDS row stride in floats (72 used; 304 B = 16B multiple)
#define LROWS 18      // halo rows (TH + 2)
#define CHUNKS 18     // float4 chunks per halo row (72 floats: x0-4 .. x0+67)
#define NBX   (IMG_W / TW)    // 16
#define NBY   (IMG_H / TH)    // 64
#define NPART (NIMG * NBY * NBX)  // 16384 partial sums
#define EPSF  1e-06f

typedef __attribute__((ext_vector_type(4))) float vf4;   // native clang vector

// ---- CDNA5 async global->LDS (ASYNCcnt), non-temporal: data is read once ---
__device__ __forceinline__ void async_ld_b128(uint32_t lds_off, const float* g) {
  // VDST = LDS byte offset (per-lane VGPR), VADDR = 64-bit global address (VGPR pair)
  asm volatile("global_load_async_to_lds_b128 %0, %1, off th:TH_LOAD_NT"
               :: "v"(lds_off), "v"(g) : "memory");
}
__device__ __forceinline__ void wait_async0() {
  asm volatile("s_wait_asynccnt 0x0" ::: "memory");
}

__global__ __launch_bounds__(256)
void refine_loss_main(const float* __restrict__ gimg,
                      const float* __restrict__ pred,
                      const float* __restrict__ mask,
                      float* __restrict__ part) {
  __shared__ __align__(16) float sg[LROWS][LW];  // grayimg halo tile
  __shared__ __align__(16) float sp[LROWS][LW];  // pred    halo tile
  __shared__ float redw[8];                      // one partial per wave32

  const int t  = threadIdx.x;
  const int bx = blockIdx.x, by = blockIdx.y, n = blockIdx.z;
  const int x0 = bx * TW, y0 = by * TH;
  const long base_n = (long)n * IMG_H * IMG_W;

  // Zero-fill only for blocks whose halo leaves the image (SAME zero padding).
  const bool border = (bx == 0) | (bx == NBX - 1) | (by == 0) | (by == NBY - 1);
  if (border) {
    float* fg = &sg[0][0];
    float* fp = &sp[0][0];
    for (int i = t; i < LROWS * LW; i += 256) { fg[i] = 0.f; fp[i] = 0.f; }
  }
  __syncthreads();

  // Stage both halo tiles via async b128 loads: rows y0-1..y0+16, cols x0-4..x0+67.
  // x0-4 keeps every 16-byte chunk aligned in both global memory and LDS.
  for (int q = t; q < 2 * LROWS * CHUNKS; q += 256) {
    const int which = q / (LROWS * CHUNKS);
    const int qq    = q - which * (LROWS * CHUNKS);
    const int r     = qq / CHUNKS;
    const int c     = qq - r * CHUNKS;
    const int gy    = y0 - 1 + r;
    const int gx    = x0 - 4 + c * 4;          // chunk fully in or fully out of image
    if ((unsigned)gy < IMG_H && (unsigned)gx < IMG_W) {
      const float* src = (which ? pred : gimg) + base_n + (long)gy * IMG_W + gx;
      uint32_t dst = (uint32_t)(uintptr_t)&(which ? sp : sg)[r][c * 4];
      async_ld_b128(dst, src);
    }
  }
  wait_async0();
  __syncthreads();

  // Each thread: 4 consecutive pixels in one row. tx in [0,16), ty in [0,16).
  const int tx = t & 15;
  const int ty = t >> 4;
  const int cb = tx * 4 + 3;                   // LDS col of (pixel x - 1); cb+1 is 16B aligned

  // Per row: 1x ds_load_b128 (cols cb+1..cb+4) + 2x ds_load_b32 (cb, cb+5).
  float a0[6], a1[6], a2[6], b0[6], b1[6], b2[6];
#pragma unroll
  for (int r = 0; r < 3; ++r) {
    float* dstA = (r == 0) ? a0 : (r == 1) ? a1 : a2;
    float* dstB = (r == 0) ? b0 : (r == 1) ? b1 : b2;
    const vf4 va = *(const vf4*)&sg[ty + r][cb + 1];
    const vf4 vb = *(const vf4*)&sp[ty + r][cb + 1];
    dstA[0] = sg[ty + r][cb];
    dstA[1] = va.x; dstA[2] = va.y; dstA[3] = va.z; dstA[4] = va.w;
    dstA[5] = sg[ty + r][cb + 5];
    dstB[0] = sp[ty + r][cb];
    dstB[1] = vb.x; dstB[2] = vb.y; dstB[3] = vb.z; dstB[4] = vb.w;
    dstB[5] = sp[ty + r][cb + 5];
  }

  // Separable stencil: column sums (1,2,1 vertical) and row diffs (top-bottom),
  // shared across the 4 output pixels.
  float cg[6], dg[6], cp[6], dp[6];
#pragma unroll
  for (int j = 0; j < 6; ++j) {
    cg[j] = fmaf(2.f, a1[j], a0[j] + a2[j]);
    dg[j] = a0[j] - a2[j];
    cp[j] = fmaf(2.f, b1[j], b0[j] + b2[j]);
    dp[j] = b0[j] - b2[j];
  }

  const vf4 mk = __builtin_nontemporal_load(
      (const vf4*)(mask + base_n + (long)(y0 + ty) * IMG_W + x0 + tx * 4));
  const float mv[4] = {mk.x, mk.y, mk.z, mk.w};

  float acc = 0.f;
#pragma unroll
  for (int j = 0; j < 4; ++j) {
    const float gx = cg[j] - cg[j + 2];
    const float gy = fmaf(2.f, dg[j + 1], dg[j] + dg[j + 2]);
    const float px = cp[j] - cp[j + 2];
    const float py = fmaf(2.f, dp[j + 1], dp[j] + dp[j + 2]);

    const float g2 = fmaf(gx, gx, fmaf(gy, gy, EPSF));
    const float p2 = fmaf(px, px, fmaf(py, py, EPSF));
    const float ig = rsqrtf(g2);
    const float ip = rsqrtf(p2);
    const float gm = g2 * ig;                  // sqrt(g2)
    const float pm = p2 * ip;                  // sqrt(p2)
    // (1 - |dot/(gm*pm)|)*pm  ==  pm - |dot|/gm
    const float dt   = fabsf(fmaf(gx, px, gy * py));
    const float lcos = pm - dt * ig;
    const float lmag = fmaxf(fmaf(1.5f, gm, -pm), 0.f);
    acc = fmaf(0.5f * (lcos + lmag), mv[j], acc);
  }

  // wave32 butterfly reduction, then one 8-entry LDS combine (deterministic).
#pragma unroll
  for (int off = 16; off > 0; off >>= 1)
    acc += __shfl_down(acc, off, 32);
  const int lane = t & 31;
  const int wave = t >> 5;
  if (lane == 0) redw[wave] = acc;
  __syncthreads();
  if (t == 0) {
    float s = 0.f;
#pragma unroll
    for (int w = 0; w < 8; ++w) s += redw[w];
    part[(n * NBY + by) * NBX + bx] = s;
  }
}

__global__ __launch_bounds__(256)
void refine_loss_reduce(const float* __restrict__ part, float* __restrict__ out) {
  __shared__ float redw[8];
  float s = 0.f;
  for (int i = threadIdx.x; i < NPART; i += 256) s += part[i];
#pragma unroll
  for (int off = 16; off > 0; off >>= 1)
    s += __shfl_down(s, off, 32);
  const int lane = threadIdx.x & 31;
  const int wave = threadIdx.x >> 5;
  if (lane == 0) redw[wave] = s;
  __syncthreads();
  if (threadIdx.x == 0) {
    float tot = 0.f;
#pragma unroll
    for (int w = 0; w < 8; ++w) tot += redw[w];
    out[0] = tot * (1.0f / ((float)NIMG * IMG_H * IMG_W));
  }
}

extern "C" void kernel_launch(void* const* d_in, const int* in_sizes, int n_in,
                              void* d_out, int out_size, void* d_ws, size_t ws_size,
                              hipStream_t stream) {
  const float* gimg = (const float*)d_in[0];
  const float* pred = (const float*)d_in[1];
  const float* mask = (const float*)d_in[2];
  float* part = (float*)d_ws;            // NPART floats = 64 KB scratch
  float* out  = (float*)d_out;

  dim3 grid(NBX, NBY, NIMG);             // 16 x 64 x 16 = 16384 blocks
  refine_loss_main<<<grid, 256, 0, stream>>>(gimg, pred, mask, part);
  refine_loss_reduce<<<1, 256, 0, stream>>>(part, out);
}